// XTermFrequency_39058432590042
// MI455X (gfx1250) — compile-verified
//
#include <hip/hip_runtime.h>
#include <stdint.h>

// Problem constants (from the reference): BATCH=512, SEQ=8192, VOCAB=50257.
#define SEQ          8192u
#define THREADS      1024u
#define VOCAB_PAD    50304u                      // counts region, padded to x4 u32
#define COUNTS_BYTES (VOCAB_PAD * 4u)            // 201216 B
#define TOKENS_BYTES (SEQ * 4u)                  // 32768 B
#define LDS_BYTES    (COUNTS_BYTES + TOKENS_BYTES) // 233984 B <= 320 KB (CDNA5 WGP LDS)

__global__ __launch_bounds__(THREADS)
void XTermFrequency_hist_kernel(const int* __restrict__ assignments,
                                const int* __restrict__ vocab_ptr,
                                float* __restrict__ out)
{
    extern __shared__ uint4 smem4[];             // [0,COUNTS_BYTES): counts u32[VOCAB_PAD]
    uint32_t* counts = (uint32_t*)smem4;         // [COUNTS_BYTES,+32KB): staged tokens
    const uint32_t tid = threadIdx.x;
    const uint32_t row = blockIdx.x;
    const int vocab = *vocab_ptr;                // uniform scalar load (50257)

    // ---- Kick off async DMA of this block's token row into LDS (ASYNCcnt) ----
    // Raw LDS byte address: dynamic LDS starts at groupstaticsize (no static LDS here).
    const uint32_t ldsDynBase = __builtin_amdgcn_groupstaticsize();
    const uint32_t tokLds     = ldsDynBase + COUNTS_BYTES + tid * 32u;   // 8 tokens/thread
    const uint64_t gAddr      = (uint64_t)(uintptr_t)(assignments
                                 + (size_t)row * SEQ + (size_t)tid * 8u);
    // ISA: the instruction offset is added to both the LDS and the global address,
    // so offset:16 advances BOTH addresses -> two b128 issues cover 32 B/lane.
    asm volatile(
        "global_load_async_to_lds_b128 %0, %1, off\n\t"
        "global_load_async_to_lds_b128 %0, %1, off offset:16"
        :
        : "v"(tokLds), "v"(gAddr)
        : "memory");

    // ---- Overlap: zero the 201 KB histogram with ds_store_b128 ----
    const uint4 z = make_uint4(0u, 0u, 0u, 0u);
    #pragma unroll 4
    for (uint32_t i = tid; i < (VOCAB_PAD / 4u); i += THREADS)
        smem4[i] = z;

    // Own-wave async loads complete -> this thread's 8 tokens are in LDS.
    asm volatile("s_wait_asynccnt 0" ::: "memory");
    __syncthreads();   // zeroing by all waves visible before atomics

    // ---- Histogram: ds_load_b128 token readback + non-returning ds_add_u32 ----
    const uint4* tok4 = (const uint4*)((const char*)smem4 + COUNTS_BYTES);
    uint4 t0 = tok4[tid * 2u + 0u];
    uint4 t1 = tok4[tid * 2u + 1u];
    atomicAdd(&counts[t0.x], 1u);
    atomicAdd(&counts[t0.y], 1u);
    atomicAdd(&counts[t0.z], 1u);
    atomicAdd(&counts[t0.w], 1u);
    atomicAdd(&counts[t1.x], 1u);
    atomicAdd(&counts[t1.y], 1u);
    atomicAdd(&counts[t1.z], 1u);
    atomicAdd(&counts[t1.w], 1u);
    __syncthreads();

    // ---- Stream out normalized row (store-bandwidth bound: ~103 MB total) ----
    // Row sum is exactly SEQ (every token hits one bucket); 1/8192 is an exact
    // power of two, so multiply == reference's divide, bit for bit.
    const float inv = 1.0f / (float)SEQ;
    float* orow = out + (size_t)row * (size_t)vocab;
    #pragma unroll 4
    for (int v = (int)tid; v < vocab; v += (int)THREADS)
        orow[v] = (float)counts[v] * inv;
}

extern "C" void kernel_launch(void* const* d_in, const int* in_sizes, int n_in,
                              void* d_out, int out_size, void* d_ws, size_t ws_size,
                              hipStream_t stream) {
    const int* assignments = (const int*)d_in[0];   // [BATCH, SEQ] int32
    const int* vocab_ptr   = (const int*)d_in[1];   // scalar vocab_size on device
    float*     out         = (float*)d_out;         // [BATCH, VOCAB] float32

    const int batch = in_sizes[0] / (int)SEQ;       // 512
    XTermFrequency_hist_kernel<<<dim3((unsigned)batch), dim3(THREADS), LDS_BYTES, stream>>>(
        assignments, vocab_ptr, out);
}